// WarpPhysicsEngineTorch_44109314129951
// MI455X (gfx1250) — compile-verified
//
#include <hip/hip_runtime.h>
#include <stdint.h>

typedef __attribute__((ext_vector_type(2))) float v2f;
typedef __attribute__((ext_vector_type(8))) float v8f;
typedef __attribute__((ext_vector_type(4))) unsigned int u32x4;
typedef __attribute__((ext_vector_type(8))) int i32x8;
typedef __attribute__((ext_vector_type(4))) int i32x4;

#define B_    4096
#define J_    4
#define P_    64
#define NB_   256
#define NPTS  512          // J*P + NB
#define HW_   1024
#define OUTC  11
#define EPS_   1e-6f
#define SIGMA_ 0.01f

__device__ __forceinline__ float clampf(float v, float lo, float hi) {
    return fminf(fmaxf(v, lo), hi);
}
__device__ __forceinline__ int clampi(int v, int lo, int hi) {
    return v < lo ? lo : (v > hi ? hi : v);
}

// --- Tensor Data Mover: DMA nelem fp32 (1-row 2D tile) global -> LDS ---
// D# layout per CDNA5 ISA sec 8.3/8.4 (group0 128b, group1 256b).
// This toolchain exposes the 6-arg builtin (g0, g1, g2, g3, extra, cpol).
__device__ __forceinline__ void tdm_load_f32(uint32_t lds_off, const void* gptr,
                                             uint32_t nelem) {
    uint64_t ga = (uint64_t)(uintptr_t)gptr;
    u32x4 g0;
    g0[0] = 1u;                                        // count=1, user mode
    g0[1] = lds_off;                                   // lds_addr [63:32]
    g0[2] = (uint32_t)(ga & 0xFFFFFFFFu);              // global_addr[31:0]
    g0[3] = (uint32_t)((ga >> 32) & 0x1FFFFFFu)        // global_addr[56:32]
          | (2u << 30);                                // type=2 ("image")
    i32x8 g1;
    g1[0] = (int)(2u << 16);                           // data_size=4B
    g1[1] = (int)((nelem & 0xFFFFu) << 16);            // tensor_dim0[15:0]
    g1[2] = (int)((nelem >> 16) | (1u << 16));         // dim0 hi | tensor_dim1=1
    g1[3] = (int)((nelem & 0xFFFFu) << 16);            // tile_dim0
    g1[4] = 1;                                         // tile_dim1=1, tile_dim2=0
    g1[5] = (int)nelem;                                // tensor_dim0_stride lo32
    g1[6] = 0;                                         // stride hi | dim1_stride lo
    g1[7] = 0;
    i32x4 z4 = {0, 0, 0, 0};                           // groups 2/3: dims 2..4 unused
    i32x8 z8 = {0, 0, 0, 0, 0, 0, 0, 0};               // extra descriptor words unused
    __builtin_amdgcn_tensor_load_to_lds(g0, g1, z4, z4, z8, 0);
}

__global__ __launch_bounds__(512) void phys_kernel(
    const float* __restrict__ x_in,     // (B,3)
    const float* __restrict__ q_in,     // (B,4)
    const float* __restrict__ thetas,   // (B,J)
    const float* __restrict__ controls, // (B,J)
    const float* __restrict__ jlp_g,    // (J,P,3)
    const float* __restrict__ jpos,     // (J,3)
    const float* __restrict__ bpts_g,   // (NB,3)
    const float* __restrict__ tdir_g,   // (J,P,3)
    const float* __restrict__ zgrid,    // (H,W)
    const float* __restrict__ zgrad,    // (2,H,W)
    float* __restrict__ out)            // (B,512,11)
{
    __shared__ float  s_jlp [J_ * P_ * 3];   // local drive points
    __shared__ float  s_tdir[J_ * P_ * 3];   // thrust directions
    __shared__ float  s_bpts[NB_ * 3];       // body points
    __shared__ float  s_pts [NPTS * 4];      // transformed points (x,y,z,pad)
    __shared__ float  s_thr [J_ * P_ * 4];   // transformed thrust
    __shared__ float4 s_out4[NPTS * OUTC / 4]; // 1408 vec4 staging for stores

    const int tid  = threadIdx.x;
    const int b    = blockIdx.x;
    const int lane = tid & 31;
    const int wave = tid >> 5;

    // ---- Phase 0: TDM constants -> LDS (wave 0 only; EXEC-uniform) ----
    if (tid < 32) {
        tdm_load_f32((uint32_t)(uintptr_t)(void*)s_jlp,  jlp_g,  J_ * P_ * 3);
        tdm_load_f32((uint32_t)(uintptr_t)(void*)s_tdir, tdir_g, J_ * P_ * 3);
        tdm_load_f32((uint32_t)(uintptr_t)(void*)s_bpts, bpts_g, NB_ * 3);
        __builtin_amdgcn_s_wait_tensorcnt(0);
    }
    __syncthreads();

    // ---- Per-block uniform state (all named scalars: no private arrays) ----
    const float qx = q_in[b * 4 + 0], qy = q_in[b * 4 + 1];
    const float qz = q_in[b * 4 + 2], qw = q_in[b * 4 + 3];
    const float Rw00 = 1.f - 2.f * (qy * qy + qz * qz);
    const float Rw01 = 2.f * (qx * qy - qz * qw);
    const float Rw02 = 2.f * (qx * qz + qy * qw);
    const float Rw10 = 2.f * (qx * qy + qz * qw);
    const float Rw11 = 1.f - 2.f * (qx * qx + qz * qz);
    const float Rw12 = 2.f * (qy * qz - qx * qw);
    const float Rw20 = 2.f * (qx * qz - qy * qw);
    const float Rw21 = 2.f * (qy * qz + qx * qw);
    const float Rw22 = 1.f - 2.f * (qx * qx + qy * qy);
    const float bx = x_in[b * 3 + 0], by = x_in[b * 3 + 1], bz = x_in[b * 3 + 2];

    // ---- Phase 1: WMMA affine transforms (3 tiles of 16 points per wave) ----
    const int row  = lane & 15;
    const int half = lane >> 4;
    #pragma unroll
    for (int i = 0; i < 3; ++i) {
        const int  g        = wave * 3 + i;      // 0..47, uniform per wave
        const bool isThrust = (g >= 32);
        const int  base     = isThrust ? (g - 32) * 16 : g * 16;
        const int  n        = lane & 15;         // point-in-tile (column)

        float M00, M01, M02, M10, M11, M12, M20, M21, M22;
        float T0, T1, T2;
        const float* lptr;
        if (!isThrust && base >= 256) {          // body points: R_world, +x
            M00 = Rw00; M01 = Rw01; M02 = Rw02;
            M10 = Rw10; M11 = Rw11; M12 = Rw12;
            M20 = Rw20; M21 = Rw21; M22 = Rw22;
            T0 = bx; T1 = by; T2 = bz;
            lptr = &s_bpts[(base - 256 + n) * 3];
        } else {                                 // drive points / thrust
            const int   j  = base >> 6;
            const float th = thetas[b * J_ + j];
            const float c = __cosf(th), s = __sinf(th);  // v_cos_f32 / v_sin_f32
            M00 = Rw00 * c - Rw02 * s;  M01 = Rw01;  M02 = Rw00 * s + Rw02 * c;
            M10 = Rw10 * c - Rw12 * s;  M11 = Rw11;  M12 = Rw10 * s + Rw12 * c;
            M20 = Rw20 * c - Rw22 * s;  M21 = Rw21;  M22 = Rw20 * s + Rw22 * c;
            if (isThrust) {
                const float vel = clampf(controls[b * J_ + j], -1.f, 1.f);
                M00 *= vel; M01 *= vel; M02 *= vel;
                M10 *= vel; M11 *= vel; M12 *= vel;
                M20 *= vel; M21 *= vel; M22 *= vel;
                T0 = T1 = T2 = 0.f;              // pure rotation*vel
                lptr = &s_tdir[(base + n) * 3];
            } else {
                const float jx = jpos[j * 3 + 0], jy = jpos[j * 3 + 1], jz = jpos[j * 3 + 2];
                T0 = Rw00 * jx + Rw01 * jy + Rw02 * jz + bx;
                T1 = Rw10 * jx + Rw11 * jy + Rw12 * jz + by;
                T2 = Rw20 * jx + Rw21 * jy + Rw22 * jz + bz;
                lptr = &s_jlp[(base + n) * 3];
            }
        }

        // Unconditional LDS reads, then pure selects (no exec-masked loads).
        const float lx = lptr[0], ly = lptr[1], lz = lptr[2];

        // A 16x4: lane=row; V0 = K0 (low half) / K2 (high half); V1 = K1 / K3.
        // Rows 0..2 = [M | t], rows 3..15 = 0 (their D rows are unused).
        const float c0 = half ? M02 : M00;
        const float c1 = half ? M12 : M10;
        const float c2 = half ? M22 : M20;
        const float d0 = half ? T0 : M01;
        const float d1 = half ? T1 : M11;
        const float d2 = half ? T2 : M21;
        const float a0 = (row == 0) ? c0 : (row == 1) ? c1 : (row == 2) ? c2 : 0.f;
        const float a1 = (row == 0) ? d0 : (row == 1) ? d1 : (row == 2) ? d2 : 0.f;
        // B 4x16: lane=column; V0 = K0/K2, V1 = K1/K3; K3 row = homogeneous 1.
        const float b0  = half ? lz   : lx;
        const float b1v = half ? 1.0f : ly;

        v2f Av = {a0, a1};
        v2f Bv = {b0, b1v};
        v8f Cv = {0.f, 0.f, 0.f, 0.f, 0.f, 0.f, 0.f, 0.f};
        v8f D = __builtin_amdgcn_wmma_f32_16x16x4_f32(
            false, Av, false, Bv, (short)0, Cv, false, false);

        // D: lanes 0..15, VGPR v = row v -> D[0..2] = (x,y,z) of column `lane`.
        if (lane < 16) {
            const int pp = base + lane;
            float* dst = isThrust ? &s_thr[pp * 4] : &s_pts[pp * 4];
            dst[0] = D[0]; dst[1] = D[1]; dst[2] = D[2];
        }
    }
    __syncthreads();

    // ---- Phase 2: terrain contact, one point per thread ----
    {
        const int t = tid;
        const float px = s_pts[t * 4 + 0];
        const float py = s_pts[t * 4 + 1];
        const float pz = s_pts[t * 4 + 2];
        float thx = 0.f, thy = 0.f, thz = 0.f;
        if (t < J_ * P_) {
            thx = s_thr[t * 4 + 0]; thy = s_thr[t * 4 + 1]; thz = s_thr[t * 4 + 2];
        }

        const float u  = clampf((px + 10.f) * 0.05f, 0.f, 1.f);
        const float v  = clampf((py + 10.f) * 0.05f, 0.f, 1.f);
        const float xg = u * (float)(HW_ - 1);
        const float yg = v * (float)(HW_ - 1);
        const int x0 = clampi((int)floorf(xg), 0, HW_ - 1);
        const int y0 = clampi((int)floorf(yg), 0, HW_ - 1);
        const int x1 = (x0 + 1 < HW_) ? x0 + 1 : HW_ - 1;
        const int y1 = (y0 + 1 < HW_) ? y0 + 1 : HW_ - 1;
        const float tx = xg - (float)x0;
        const float ty = yg - (float)y0;

        auto bil = [&](const float* grid) -> float {
            const float c00 = grid[y0 * HW_ + x0];
            const float c10 = grid[y0 * HW_ + x1];
            const float c01 = grid[y1 * HW_ + x0];
            const float c11 = grid[y1 * HW_ + x1];
            const float a  = c00 * (1.f - tx) + c10 * tx;
            const float b2 = c01 * (1.f - tx) + c11 * tx;
            return a * (1.f - ty) + b2 * ty;
        };
        const float zt = bil(zgrid);
        const float gx = bil(zgrad);
        const float gy = bil(zgrad + HW_ * HW_);

        // n = (-gx,-gy,1)/nrm; nrm = sqrt(gx^2+gy^2+1) >= 1, so nrm>EPS <=> s2>EPS^2.
        const float s2 = gx * gx + gy * gy + 1.f;
        float nx, ny, nz;
        if (s2 > EPS_ * EPS_) {
            const float inv = rsqrtf(s2);        // v_rsq_f32
            nx = -gx * inv; ny = -gy * inv; nz = inv;
        } else {
            nx = ny = nz = 0.f;
        }
        const float dh = (pz - zt) * nz;
        // 0.5*(1+tanh(y)) == 1/(1+exp(-2y)); y = dh/(SIGMA+EPS)
        const float contact = (dh > 0.f)
            ? 1.f / (1.f + __expf(dh * (-2.f / (SIGMA_ + EPS_)))) : 0.f;
        const float dhp = (dh > 0.f ? dh : 0.f) * contact;

        float* o = ((float*)s_out4) + t * OUTC;
        o[0] = px;  o[1] = py;  o[2] = pz;
        o[3] = thx; o[4] = thy; o[5] = thz;
        o[6] = nx;  o[7] = ny;  o[8] = nz;
        o[9] = dhp; o[10] = contact;
    }
    __syncthreads();

    // ---- Phase 3: streaming b128 writeback (store-bandwidth bound) ----
    float4* outv = (float4*)(out + (size_t)b * (NPTS * OUTC));
    #pragma unroll
    for (int i = tid; i < NPTS * OUTC / 4; i += 512) {
        outv[i] = s_out4[i];
    }
}

extern "C" void kernel_launch(void* const* d_in, const int* in_sizes, int n_in,
                              void* d_out, int out_size, void* d_ws, size_t ws_size,
                              hipStream_t stream) {
    (void)in_sizes; (void)n_in; (void)out_size; (void)d_ws; (void)ws_size;
    const float* x    = (const float*)d_in[0];
    const float* q    = (const float*)d_in[2];
    const float* th   = (const float*)d_in[4];
    const float* ctl  = (const float*)d_in[5];
    const float* jlp  = (const float*)d_in[6];
    const float* jp   = (const float*)d_in[7];
    const float* bp   = (const float*)d_in[13];
    const float* td   = (const float*)d_in[14];
    const float* zg   = (const float*)d_in[15];
    const float* zgr  = (const float*)d_in[16];
    phys_kernel<<<B_, 512, 0, stream>>>(x, q, th, ctl, jlp, jp, bp, td, zg, zgr,
                                        (float*)d_out);
}